// GNNEmbedding_42374147342459
// MI455X (gfx1250) — compile-verified
//
#include <hip/hip_runtime.h>
#include <math.h>

// GNN embedding, fully fused per batch element on gfx1250.
// fp32 WMMA (V_WMMA_F32_16X16X4_F32) for all GEMMs; LDS-resident intermediates.

#define NEL 16
#define NION 4

typedef float v2f __attribute__((ext_vector_type(2)));
typedef float v8f __attribute__((ext_vector_type(8)));

__device__ __attribute__((always_inline)) inline v8f wmma4(v2f a, v2f b, v8f c) {
  // D = A(16x4 f32) * B(4x16 f32) + C(16x16 f32)
  return __builtin_amdgcn_wmma_f32_16x16x4_f32(false, a, false, b, (short)0, c,
                                               false, false);
}

// A fragment: rows on lanes (m = lane&15), K split: VGPR0/1 hold K = k0+2*half .. +1
__device__ __attribute__((always_inline)) inline v2f ldA(const float* A, int lda,
                                                         int k0, int lane) {
  const int m = lane & 15, h = lane >> 4;
  const float* p = A + m * lda + k0 + 2 * h;
  v2f a; a.x = p[0]; a.y = p[1];
  return a;
}

// B fragment: cols on lanes (n = lane&15), rows K = k0+2*half, k0+2*half+1
__device__ __attribute__((always_inline)) inline v2f ldB(const float* W, int ldb,
                                                         int k0, int n0, int lane) {
  const int n = lane & 15, h = lane >> 4;
  v2f b;
  b.x = W[(k0 + 2 * h) * ldb + n0 + n];
  b.y = W[(k0 + 2 * h + 1) * ldb + n0 + n];
  return b;
}

__device__ __attribute__((always_inline)) inline v8f bsplat(const float* bias,
                                                            int n0, int lane) {
  const float v = bias[n0 + (lane & 15)];
  v8f c;
#pragma unroll
  for (int r = 0; r < 8; ++r) c[r] = v;
  return c;
}

__device__ __attribute__((always_inline)) inline float fast_tanh(float x) {
  x = fminf(15.0f, fmaxf(-15.0f, x));
  const float e = __expf(2.0f * x);
  return (e - 1.0f) / (e + 1.0f);
}

// store 16x16 tile with tanh; row m = r + 8*half, col n = lane&15
__device__ __attribute__((always_inline)) inline void st_tanh(float* D, int ldd,
                                                              v8f c, int lane) {
  const int n = lane & 15, h = lane >> 4;
#pragma unroll
  for (int r = 0; r < 8; ++r) D[(r + 8 * h) * ldd + n] = fast_tanh(c[r]);
}

__device__ __attribute__((always_inline)) inline v8f gemm_tile(const float* A, int lda,
                                                               const float* W, int ldb,
                                                               int n0, int K, v8f c,
                                                               int lane) {
  for (int k = 0; k < K; k += 4)
    c = wmma4(ldA(A, lda, k, lane), ldB(W, ldb, k, n0, lane), c);
  return c;
}

struct Params {
  const float *feat_el, *feat_el_el, *feat_el_ion, *feat_ion;
  const float *dist_el_el, *dist_el_ion;
  const float *Wei0, *bei0, *Wei1, *bei1;
  const float *Wsa0, *bsa0, *Wsa1, *bsa1;
  const float *Wdi0, *bdi0, *Wdi1, *bdi1;
  const float *Ws_ei, *bs_ei, *Wu_ei, *bu_ei, *We_ei, *be_ei;
  const float *Ws0, *bs0, *Wu0, *bu0, *We0, *be0;
  const float *Ws1, *bs1, *Wu1, *bu1, *We1, *be1;
  float *out_emb, *out_ion, *out_ee, *out_ei;
};

__global__ __launch_bounds__(256) void gnn_fused_kernel(Params p) {
  const int b = blockIdx.x;
  const int tid = threadIdx.x;
  const int lane = tid & 31;
  const int wave = tid >> 5;

  // ---- LDS (~210 KB of the 320 KB WGP LDS) ----
  __shared__ float sE0[320 * 4];          // routed raw edge feats (rows x 4)
  __shared__ float sStage[8][16 * 64];    // per-wave layer0 staging
  __shared__ float sEE[NEL * NEL * 64];   // el-el edge feats e0 (natural i,j)
  __shared__ float sEE2[NEL * NEL * 64];  // el-el edge feats e1 (after iter0)
  __shared__ float sEI[NEL * NION * 64];  // el-ion edge feats
  __shared__ float sX[NEL * 32];          // feat_el
  __shared__ float sEmb[NEL * 128];       // emb after el-ion MP
  __shared__ float sEmb2[NEL * 128];      // emb after iter0
  __shared__ float sAgg[NEL * 64];        // aggregated messages
  __shared__ float sS[NEL * 64];          // projected senders
  __shared__ float sSei[NION * 64];       // projected ion senders
  __shared__ float sMee[NEL * NEL];
  __shared__ float sMei[NEL * NION];

  if (b == 0 && tid < NION * 32) p.out_ion[tid] = p.feat_ion[tid];

  // ---- phase 0: stage inputs, masks, ion sender projection ----
  for (int i = tid; i < NEL * 32; i += 256) sX[i] = p.feat_el[b * NEL * 32 + i];
  {
    const int pi = tid;                    // 256 el-el pairs, one per thread
    const int i = pi >> 4, j = pi & 15;
    int r;                                 // route by spin block
    if (i < 8) r = (j < 8) ? (i * 8 + j) : (128 + i * 8 + (j - 8));
    else       r = (j < 8) ? (192 + (i - 8) * 8 + j) : (64 + (i - 8) * 8 + (j - 8));
#pragma unroll
    for (int f = 0; f < 4; ++f)
      sE0[r * 4 + f] = p.feat_el_el[(b * 256 + pi) * 4 + f];
    sMee[pi] = __expf(p.dist_el_el[b * 256 + pi] * -0.2f);
  }
  if (tid < NEL * NION) {
#pragma unroll
    for (int f = 0; f < 4; ++f)
      sE0[(256 + tid) * 4 + f] = p.feat_el_ion[(b * 64 + tid) * 4 + f];
    sMei[tid] = __expf(p.dist_el_ion[b * 64 + tid] * -0.2f);
  }
  {
    const int j = tid >> 6, nn = tid & 63; // s_ion = tanh(feat_ion @ Ws_ei + b)
    float acc = p.bs_ei[nn];
#pragma unroll
    for (int k = 0; k < 32; ++k) acc += p.feat_ion[j * 32 + k] * p.Ws_ei[k * 64 + nn];
    sSei[tid] = fast_tanh(acc);
  }
  __syncthreads();

  // ---- phase 1: edge MLPs (20 row-tiles: 8 same, 8 diff, 4 el-ion) ----
  for (int rt = wave; rt < 20; rt += 8) {
    const float *W0, *B0, *W1, *B1;
    if (rt < 8)       { W0 = p.Wsa0; B0 = p.bsa0; W1 = p.Wsa1; B1 = p.bsa1; }
    else if (rt < 16) { W0 = p.Wdi0; B0 = p.bdi0; W1 = p.Wdi1; B1 = p.bdi1; }
    else              { W0 = p.Wei0; B0 = p.bei0; W1 = p.Wei1; B1 = p.bei1; }
    const float* A0 = sE0 + rt * 16 * 4;
    float* stg = sStage[wave];
#pragma unroll
    for (int ct = 0; ct < 4; ++ct) {       // layer 0: K=4, single WMMA
      v8f c = bsplat(B0, ct * 16, lane);
      c = wmma4(ldA(A0, 4, 0, lane), ldB(W0, 64, 0, ct * 16, lane), c);
      st_tanh(stg + ct * 16, 64, c, lane);
    }
#pragma unroll
    for (int ct = 0; ct < 4; ++ct) {       // layer 1: K=64
      v8f c = bsplat(B1, ct * 16, lane);
      c = gemm_tile(stg, 64, W1, 64, ct * 16, 64, c, lane);
      const int n = lane & 15, h = lane >> 4;
#pragma unroll
      for (int r = 0; r < 8; ++r) {        // scatter routed row -> natural (i,j)
        const int rr = rt * 16 + r + 8 * h;
        const float val = fast_tanh(c[r]);
        if (rr < 256) {
          const int g = rr >> 6, loc = rr & 63, a = loc >> 3, q = loc & 7;
          const int i = (g == 0 || g == 2) ? a : a + 8;
          const int j = (g == 0 || g == 3) ? q : q + 8;
          sEE[(i * 16 + j) * 64 + ct * 16 + n] = val;
        } else {
          sEI[(rr - 256) * 64 + ct * 16 + n] = val;
        }
      }
    }
  }
  __syncthreads();

  // ---- phase 2: el-ion aggregation ----
  for (int idx = tid; idx < NEL * 64; idx += 256) {
    const int i = idx >> 6, nn = idx & 63;
    float acc = 0.f;
#pragma unroll
    for (int j = 0; j < NION; ++j)
      acc += sEI[(i * NION + j) * 64 + nn] * sSei[j * 64 + nn] * sMei[i * NION + j];
    sAgg[idx] = acc;
  }
  __syncthreads();

  // ---- phase 3: node update [feat_el | agg] @ Wu_ei, + el-ion edge output ----
  {
    const int ct = wave;                   // 8 col tiles of D_NODE=128
    v8f c = bsplat(p.bu_ei, ct * 16, lane);
    for (int k = 0; k < 32; k += 4)
      c = wmma4(ldA(sX, 32, k, lane), ldB(p.Wu_ei, 128, k, ct * 16, lane), c);
    for (int k = 0; k < 64; k += 4)
      c = wmma4(ldA(sAgg, 64, k, lane), ldB(p.Wu_ei, 128, 32 + k, ct * 16, lane), c);
    st_tanh(sEmb + ct * 16, 128, c, lane);
  }
  for (int t = wave; t < 16; t += 8) {     // edge_el_ion out = tanh(e @ We_ei + be)
    const int rt = t >> 2, ct = t & 3;
    v8f c = bsplat(p.be_ei, ct * 16, lane);
    c = gemm_tile(sEI + rt * 16 * 64, 64, p.We_ei, 64, ct * 16, 64, c, lane);
    st_tanh(p.out_ei + (size_t)b * 64 * 64 + rt * 16 * 64 + ct * 16, 64, c, lane);
  }
  __syncthreads();

  // ---- iteration 0 ----
  if (wave < 4) {                          // s0 = tanh(emb @ Ws0 + bs0), K=128
    const int ct = wave;
    v8f c = bsplat(p.bs0, ct * 16, lane);
    c = gemm_tile(sEmb, 128, p.Ws0, 64, ct * 16, 128, c, lane);
    st_tanh(sS + ct * 16, 64, c, lane);
  }
  __syncthreads();
  for (int idx = tid; idx < NEL * 64; idx += 256) {
    const int i = idx >> 6, nn = idx & 63;
    float acc = 0.f;
#pragma unroll
    for (int j = 0; j < NEL; ++j)
      acc += sEE[(i * 16 + j) * 64 + nn] * sS[j * 64 + nn] * sMee[i * 16 + j];
    sAgg[idx] = acc;
  }
  __syncthreads();
  {                                        // emb2 = tanh([emb | agg] @ Wu0 + bu0)
    const int ct = wave;
    v8f c = bsplat(p.bu0, ct * 16, lane);
    for (int k = 0; k < 128; k += 4)
      c = wmma4(ldA(sEmb, 128, k, lane), ldB(p.Wu0, 128, k, ct * 16, lane), c);
    for (int k = 0; k < 64; k += 4)
      c = wmma4(ldA(sAgg, 64, k, lane), ldB(p.Wu0, 128, 128 + k, ct * 16, lane), c);
    st_tanh(sEmb2 + ct * 16, 128, c, lane);
  }
  for (int t = wave; t < 64; t += 8) {     // e1 = tanh(e0 @ We0 + be0)
    const int rt = t >> 2, ct = t & 3;
    v8f c = bsplat(p.be0, ct * 16, lane);
    c = gemm_tile(sEE + rt * 16 * 64, 64, p.We0, 64, ct * 16, 64, c, lane);
    st_tanh(sEE2 + rt * 16 * 64 + ct * 16, 64, c, lane);
  }
  __syncthreads();

  // ---- iteration 1 ----
  if (wave < 4) {
    const int ct = wave;
    v8f c = bsplat(p.bs1, ct * 16, lane);
    c = gemm_tile(sEmb2, 128, p.Ws1, 64, ct * 16, 128, c, lane);
    st_tanh(sS + ct * 16, 64, c, lane);
  }
  __syncthreads();
  for (int idx = tid; idx < NEL * 64; idx += 256) {
    const int i = idx >> 6, nn = idx & 63;
    float acc = 0.f;
#pragma unroll
    for (int j = 0; j < NEL; ++j)
      acc += sEE2[(i * 16 + j) * 64 + nn] * sS[j * 64 + nn] * sMee[i * 16 + j];
    sAgg[idx] = acc;
  }
  __syncthreads();
  {                                        // final emb -> global
    const int ct = wave;
    v8f c = bsplat(p.bu1, ct * 16, lane);
    for (int k = 0; k < 128; k += 4)
      c = wmma4(ldA(sEmb2, 128, k, lane), ldB(p.Wu1, 128, k, ct * 16, lane), c);
    for (int k = 0; k < 64; k += 4)
      c = wmma4(ldA(sAgg, 64, k, lane), ldB(p.Wu1, 128, 128 + k, ct * 16, lane), c);
    st_tanh(p.out_emb + (size_t)b * 16 * 128 + ct * 16, 128, c, lane);
  }
  for (int t = wave; t < 64; t += 8) {     // final el-el edges -> global
    const int rt = t >> 2, ct = t & 3;
    v8f c = bsplat(p.be1, ct * 16, lane);
    c = gemm_tile(sEE2 + rt * 16 * 64, 64, p.We1, 64, ct * 16, 64, c, lane);
    st_tanh(p.out_ee + (size_t)b * 16 * 16 * 64 + rt * 16 * 64 + ct * 16, 64, c, lane);
  }
}

extern "C" void kernel_launch(void* const* d_in, const int* in_sizes, int n_in,
                              void* d_out, int out_size, void* d_ws, size_t ws_size,
                              hipStream_t stream) {
  (void)n_in; (void)d_ws; (void)ws_size; (void)out_size;
  Params p;
  p.feat_el     = (const float*)d_in[0];
  p.feat_el_el  = (const float*)d_in[1];
  p.feat_el_ion = (const float*)d_in[2];
  p.feat_ion    = (const float*)d_in[3];
  // d_in[4] feat_ion_ion unused (ion_gnn='none')
  p.dist_el_el  = (const float*)d_in[5];
  p.dist_el_ion = (const float*)d_in[6];
  p.Wei0 = (const float*)d_in[7];  p.bei0 = (const float*)d_in[8];
  p.Wei1 = (const float*)d_in[9];  p.bei1 = (const float*)d_in[10];
  p.Wsa0 = (const float*)d_in[11]; p.bsa0 = (const float*)d_in[12];
  p.Wsa1 = (const float*)d_in[13]; p.bsa1 = (const float*)d_in[14];
  p.Wdi0 = (const float*)d_in[15]; p.bdi0 = (const float*)d_in[16];
  p.Wdi1 = (const float*)d_in[17]; p.bdi1 = (const float*)d_in[18];
  // d_in[19..22] Wio/bio unused
  p.Ws_ei = (const float*)d_in[23]; p.bs_ei = (const float*)d_in[24];
  p.Wu_ei = (const float*)d_in[25]; p.bu_ei = (const float*)d_in[26];
  p.We_ei = (const float*)d_in[27]; p.be_ei = (const float*)d_in[28];
  p.Ws0 = (const float*)d_in[29]; p.bs0 = (const float*)d_in[30];
  p.Wu0 = (const float*)d_in[31]; p.bu0 = (const float*)d_in[32];
  p.We0 = (const float*)d_in[33]; p.be0 = (const float*)d_in[34];
  p.Ws1 = (const float*)d_in[35]; p.bs1 = (const float*)d_in[36];
  p.Wu1 = (const float*)d_in[37]; p.bu1 = (const float*)d_in[38];
  p.We1 = (const float*)d_in[39]; p.be1 = (const float*)d_in[40];
  // d_in[41] n_up (fixed 8)

  const int Bn = in_sizes[0] / (NEL * 32);
  float* out = (float*)d_out;
  p.out_emb = out;                                       // [B,16,128]
  p.out_ion = out + (size_t)Bn * NEL * 128;              // [4,32]
  p.out_ee  = p.out_ion + NION * 32;                     // [B,16,16,64]
  p.out_ei  = p.out_ee + (size_t)Bn * NEL * NEL * 64;    // [B,16,4,64]

  gnn_fused_kernel<<<dim3(Bn), dim3(256), 0, stream>>>(p);
}